// MultiHeadedAttention_3_18631568130077
// MI455X (gfx1250) — compile-verified
//
#include <hip/hip_runtime.h>
#include <hip/hip_bf16.h>

// Problem constants (match setup_inputs)
#define B_   8
#define F_   256
#define C_   64
#define H_   8
#define K2_  128          // 2*C
#define KSEL 32           // node_num
#define WPAD 132          // padded row stride for W tile in LDS (floats)
#define APAD 68           // padded row stride for A tile in LDS (floats)

typedef __attribute__((ext_vector_type(2))) float v2f;
typedef __attribute__((ext_vector_type(8))) float v8f;

// ---------------------------------------------------------------------------
// Kernel 1: fused  concat -> GEMM(2C x H) -> +bias -> sigmoid -> *roi
// One block per (b,i).  8 waves; each wave computes two 16(j) x 16(h) tiles
// with V_WMMA_F32_16X16X4_F32, K=128 in 32 steps (query half, then key half).
// Writes A1 = sigmoid(x)*roi  (the tensor jax top_k ranks on).
// ---------------------------------------------------------------------------
__global__ __launch_bounds__(256) void fused_attn_gemm(
    const float* __restrict__ q, const float* __restrict__ kk,
    const float* __restrict__ roi, const float* __restrict__ W,
    const float* __restrict__ bias, float* __restrict__ out)
{
    __shared__ float sW[16 * WPAD];        // W padded to 16 heads (8..15 = 0)
    __shared__ float sA[8][16 * APAD];     // per-wave 16 x 64 A half-tile

    const int tid  = threadIdx.x;
    const int lane = tid & 31;
    const int wave = tid >> 5;
    const int bi   = blockIdx.x;           // b*F + i

    // Stage W -> LDS (zero-padded heads 8..15), cooperatively.
    for (int idx = tid; idx < 16 * K2_; idx += 256) {
        const int n = idx >> 7;            // head row
        const int c = idx & 127;           // k column
        sW[n * WPAD + c] = (n < H_) ? W[n * K2_ + c] : 0.0f;
    }
    __syncthreads();

    const int   hsel = lane & 15;                    // N (head) column
    const float bv   = (hsel < H_) ? bias[hsel] : 0.0f;
    const float* qbase = q   + (size_t)bi * F_ * C_;
    const float* kbase = kk  + (size_t)bi * F_ * C_;
    const float* rbase = roi + (size_t)bi * F_;
    float*       obase = out + (size_t)bi * F_ * H_;

    const int rowHalf = lane >> 4;                   // staging: row parity
    const int cg      = (lane & 15) * 4;             // staging: 16B col group
    // WMMA f32 16x16x4 fragment addressing (ISA 7.12.2):
    //   A: lanes 0-15 -> (M=lane,   K=k0,k0+1); lanes 16-31 -> (M=lane-16, K=k0+2,k0+3)
    //   B: lanes 0-15 -> (K=k0,k0+1, N=lane);   lanes 16-31 -> (K=k0+2,k0+3, N=lane-16)
    const int aofs = (lane & 15) * APAD + (lane >> 4) * 2;
    const int bofs = (lane & 15) * WPAD + (lane >> 4) * 2;

    for (int jt = wave; jt < 16; jt += 8) {          // 2 uniform iterations/wave
        const int j0 = jt * 16;
        v8f acc = {};
        for (int ph = 0; ph < 2; ++ph) {             // 0: query half, 1: key half
            const float* src = ph ? kbase : qbase;
            // Coalesced b128 staging of this wave's 16x64 tile.
            #pragma unroll
            for (int t = 0; t < 8; ++t) {
                const int r = t * 2 + rowHalf;
                const float4 v4 = *(const float4*)(src + (size_t)(j0 + r) * C_ + cg);
                *(float4*)(&sA[wave][r * APAD + cg]) = v4;
            }
            if (ph == 0) {   // pull the key half toward L2 while we compute
                __builtin_prefetch(kbase + (size_t)(j0 + (lane & 15)) * C_, 0, 1);
            }
            __syncthreads();
            const int bph = bofs + ph * C_;          // key half uses W cols 64..127
            #pragma unroll
            for (int s = 0; s < 16; ++s) {           // 16 K-steps of 4
                v2f a = *(const v2f*)(&sA[wave][aofs + s * 4]);
                v2f b = *(const v2f*)(&sW[bph + s * 4]);
                acc = __builtin_amdgcn_wmma_f32_16x16x4_f32(
                          false, a, false, b, (short)0, acc, false, false);
            }
            __syncthreads();
        }
        // Epilogue: D[M=vr + 8*(lane>=16)][N=lane&15]
        if (hsel < H_) {
            const int mofs = (lane >> 4) * 8;
            #pragma unroll
            for (int vr = 0; vr < 8; ++vr) {
                const int j = j0 + vr + mofs;
                const float x  = acc[vr] + bv;
                const float sg = 1.0f / (1.0f + __expf(-x));
                obase[(size_t)j * H_ + hsel] = sg * rbase[j];
            }
        }
    }
}

// ---------------------------------------------------------------------------
// Kernel 2: zero the union-mask (ws is poisoned by the harness).
// ---------------------------------------------------------------------------
__global__ void zero_present(float* __restrict__ present)
{
    present[threadIdx.x] = 0.0f;
}

// ---------------------------------------------------------------------------
// Kernel 3: per-(b,i,h) top-32 of 256 along j; union of indices -> present[].
// One wave32 per problem; 8 values/lane; 32 rounds of butterfly arg-max with
// jax-compatible tie-break (equal value -> lower index).
// ---------------------------------------------------------------------------
__global__ __launch_bounds__(256) void topk_union(
    const float* __restrict__ attn, float* __restrict__ present)
{
    const int lane = threadIdx.x & 31;
    const int wave = threadIdx.x >> 5;
    const int p    = blockIdx.x * 8 + wave;        // (b,i,h) problem id
    const int b    = p / (F_ * H_);
    const int rem  = p % (F_ * H_);
    const int i    = rem / H_;
    const int h    = rem % H_;
    const float* base = attn + (size_t)(b * F_ + i) * F_ * H_ + h;

    float vals[8];
    #pragma unroll
    for (int t = 0; t < 8; ++t)
        vals[t] = base[(size_t)(lane + t * 32) * H_];   // all >= 0

    for (int r = 0; r < KSEL; ++r) {
        float bv = -1.0f; int bj = 0x7fffffff;
        #pragma unroll
        for (int t = 0; t < 8; ++t) {
            const int j = lane + t * 32;
            if (vals[t] > bv) { bv = vals[t]; bj = j; }   // ascending t -> low j wins ties
        }
        #pragma unroll
        for (int off = 16; off > 0; off >>= 1) {
            const float ov = __shfl_xor(bv, off, 32);
            const int   oj = __shfl_xor(bj, off, 32);
            if (ov > bv || (ov == bv && oj < bj)) { bv = ov; bj = oj; }
        }
        if ((bj & 31) == lane) vals[bj >> 5] = -1.0f;     // remove the winner
        if (lane == 0) present[bj] = 1.0f;                // benign races (all write 1)
    }
}

// ---------------------------------------------------------------------------
// Kernel 4: out = A1 * roi * present[j]  ==  sigmoid * roi^2 * present.
// Exactly B*F*F*H = 16384*256 elements.
// ---------------------------------------------------------------------------
__global__ __launch_bounds__(256) void finalize(
    float* __restrict__ out, const float* __restrict__ roi,
    const float* __restrict__ present)
{
    const size_t idx = (size_t)blockIdx.x * 256 + threadIdx.x;
    const int    h   = (int)(idx & (H_ - 1)); (void)h;
    const size_t r   = idx >> 3;              // / H
    const int    j   = (int)(r & (F_ - 1));
    const size_t bi  = r >> 8;                // / F  -> b*F + i
    out[idx] *= roi[bi * F_ + j] * present[j];
}

// ---------------------------------------------------------------------------
extern "C" void kernel_launch(void* const* d_in, const int* in_sizes, int n_in,
                              void* d_out, int out_size, void* d_ws, size_t ws_size,
                              hipStream_t stream)
{
    (void)in_sizes; (void)n_in; (void)out_size; (void)ws_size;
    const float* q    = (const float*)d_in[0];
    const float* k    = (const float*)d_in[1];
    const float* roi  = (const float*)d_in[2];
    const float* W    = (const float*)d_in[3];
    const float* bias = (const float*)d_in[4];
    // d_in[5] = node_num (compile-time constant 32 here)
    float* out     = (float*)d_out;
    float* present = (float*)d_ws;            // 256 floats of scratch

    zero_present  <<<1,                     256, 0, stream>>>(present);
    fused_attn_gemm<<<B_ * F_,              256, 0, stream>>>(q, k, roi, W, bias, out);
    topk_union    <<<(B_ * F_ * H_) / 8,    256, 0, stream>>>(out, present);
    finalize      <<<(B_ * F_ * F_ * H_) / 256, 256, 0, stream>>>(out, roi, present);
}